// GraphAttn_52424370815603
// MI455X (gfx1250) — compile-verified
//
#include <hip/hip_runtime.h>
#include <hip/hip_bf16.h>

// Problem constants (match reference setup_inputs)
#define N_ROWS 8192
#define IN_DIM 512
#define OUT_DIM 256
#define ALPHA 0.2f

typedef __attribute__((ext_vector_type(2))) float v2f;
typedef __attribute__((ext_vector_type(8))) float v8f;

// D = A(16x4 f32) * B(4x16 f32) + C(16x16 f32), full wave32 op.
__device__ __forceinline__ v8f wmma_f32(v2f a, v2f b, v8f c) {
    // 8 args: (neg_a, A, neg_b, B, c_mod, C, reuse_a, reuse_b)
    return __builtin_amdgcn_wmma_f32_16x16x4_f32(false, a, false, b, (short)0, c,
                                                 false, false);
}

__device__ __forceinline__ float leaky(float e) { return e > 0.0f ? e : ALPHA * e; }

// ---------------------------------------------------------------------------
// Phase A: h[N, OUT] = x[N, IN] @ W[OUT, IN]^T     (f32 WMMA 16x16x4)
// Block = 256 threads = 8 waves. Wave w: i-tile = blk*2 + (w>>2), o-base = (w&3)*64.
// Each wave computes a 16x64 output tile (4 accumulator tiles), K-loop of 128.
// ---------------------------------------------------------------------------
__global__ __launch_bounds__(256) void gat_gemm_h(const float* __restrict__ x,
                                                  const float* __restrict__ W,
                                                  float* __restrict__ h) {
    const int lane = threadIdx.x & 31;
    const int wave = threadIdx.x >> 5;
    const int m    = lane & 15;   // A row / B col within tile
    const int half = lane >> 4;   // K half: lanes 0-15 -> K{0,1}, 16-31 -> K{2,3}
    const int i0   = (blockIdx.x * 2 + (wave >> 2)) * 16;
    const int ob   = (wave & 3) * 64;

    v8f c[4] = {};
    const float* xrow = x + (size_t)(i0 + m) * IN_DIM + 2 * half;

    for (int k = 0; k < IN_DIM; k += 4) {
        v2f a = *(const v2f*)(xrow + k);  // x[i0+m, k + 2*half + {0,1}]
#pragma unroll
        for (int t = 0; t < 4; ++t) {
            // B[kk, n] = W[ob + t*16 + n, k + kk]; lane holds n=m, kk = 2*half+{0,1}
            v2f b = *(const v2f*)(W + (size_t)(ob + t * 16 + m) * IN_DIM + k + 2 * half);
            c[t] = wmma_f32(a, b, c[t]);
        }
    }
#pragma unroll
    for (int t = 0; t < 4; ++t)
#pragma unroll
        for (int r = 0; r < 8; ++r)
            h[(size_t)(i0 + r + 8 * half) * OUT_DIM + ob + t * 16 + m] = c[t][r];
}

// ---------------------------------------------------------------------------
// Phase B1: s1 = h @ a1, s2 = h @ a2. One wave per row.
// ---------------------------------------------------------------------------
__global__ __launch_bounds__(256) void gat_s12(const float* __restrict__ h,
                                               const float* __restrict__ a1,
                                               const float* __restrict__ a2,
                                               float* __restrict__ s1,
                                               float* __restrict__ s2) {
    const int lane = threadIdx.x & 31;
    const int row  = blockIdx.x * 8 + (threadIdx.x >> 5);
    const float* hr = h + (size_t)row * OUT_DIM;
    float acc1 = 0.0f, acc2 = 0.0f;
#pragma unroll
    for (int t = 0; t < OUT_DIM / 32; ++t) {
        float hv = hr[lane + 32 * t];
        acc1 = fmaf(hv, a1[lane + 32 * t], acc1);
        acc2 = fmaf(hv, a2[lane + 32 * t], acc2);
    }
#pragma unroll
    for (int off = 16; off > 0; off >>= 1) {
        acc1 += __shfl_down(acc1, off, 32);
        acc2 += __shfl_down(acc2, off, 32);
    }
    if (lane == 0) { s1[row] = acc1; s2[row] = acc2; }
}

// ---------------------------------------------------------------------------
// Phase B2: s1max = max(s1). Single block reduction.
// ---------------------------------------------------------------------------
__global__ __launch_bounds__(256) void gat_s1max(const float* __restrict__ s1,
                                                 float* __restrict__ s1max) {
    __shared__ float red[256];
    float m = -3.402823466e38f;
    for (int i = threadIdx.x; i < N_ROWS; i += 256) m = fmaxf(m, s1[i]);
    red[threadIdx.x] = m;
    __syncthreads();
    for (int s = 128; s > 0; s >>= 1) {
        if ((int)threadIdx.x < s) red[threadIdx.x] = fmaxf(red[threadIdx.x], red[threadIdx.x + s]);
        __syncthreads();
    }
    if (threadIdx.x == 0) *s1max = red[0];
}

// ---------------------------------------------------------------------------
// Phase C: out = softmax_row(leaky(s2[i]+s1[j])) @ h, fused flash-style.
// Since leaky() is monotonic, row max m_i = leaky(s2[i] + max_j s1[j]) exactly.
// Block = 8 waves; wave w owns i-tile (16 rows) x all 256 cols (16 C tiles).
// h is staged in 32-row chunks into LDS *transposed* (hs2[col][j], stride 34
// floats -> 8B-aligned ds_load_b64 of the K-pair, conflict-free banks).
// P-tiles (A matrix) are computed in registers: 2 exp per lane per K-step.
// ---------------------------------------------------------------------------
#define JC 32
#define HSTRIDE 34   // even (b64 alignment) + bank-conflict-free padding

__global__ __launch_bounds__(256) void gat_attn(const float* __restrict__ h,
                                                const float* __restrict__ s1,
                                                const float* __restrict__ s2,
                                                const float* __restrict__ s1maxp,
                                                float* __restrict__ out) {
    __shared__ float hs2[OUT_DIM * HSTRIDE];  // hs2[col*HSTRIDE + j] = h[jc+j, col]
    __shared__ float s1s[JC];

    const int lane = threadIdx.x & 31;
    const int wave = threadIdx.x >> 5;
    const int m    = lane & 15;
    const int half = lane >> 4;
    const int i0   = blockIdx.x * 128 + wave * 16;

    const float s1max = *s1maxp;
    const float s2i   = s2[i0 + m];          // per-lane row constant
    const float mi    = leaky(s2i + s1max);  // exact row max of scores

    v8f  c[16] = {};
    float z = 0.0f;  // per-lane partial of row-sum Z

    for (int jc = 0; jc < N_ROWS; jc += JC) {
        // ---- cooperative stage: h[jc..jc+31, :] -> transposed LDS ----
        const float* hsrc = h + (size_t)jc * OUT_DIM;
#pragma unroll 4
        for (int it = 0; it < JC; ++it)  // threads 0..255 read one full h row
            hs2[threadIdx.x * HSTRIDE + it] = hsrc[(size_t)it * OUT_DIM + threadIdx.x];
        if (threadIdx.x < JC) s1s[threadIdx.x] = s1[jc + threadIdx.x];
        __syncthreads();

        // ---- 8 K-steps of 4, 16 WMMAs each ----
#pragma unroll
        for (int js = 0; js < JC; js += 4) {
            const int j0 = js + 2 * half;  // this lane's K pair within the step
            float p0 = __expf(leaky(s2i + s1s[j0])     - mi);
            float p1 = __expf(leaky(s2i + s1s[j0 + 1]) - mi);
            z += p0 + p1;
            v2f a = {p0, p1};  // A[m, j0], A[m, j0+1]
#pragma unroll
            for (int t = 0; t < 16; ++t) {
                // B[kk, n] = h[jc + j0 + {0,1}, t*16 + n]; one b64 LDS read
                v2f b = *(const v2f*)&hs2[(t * 16 + m) * HSTRIDE + j0];
                c[t] = wmma_f32(a, b, c[t]);
            }
        }
        __syncthreads();
    }

    // ---- Z: combine the two K-halves; lane L holds Z for row i0 + (L&15) ----
    float zfull = z + __shfl_xor(z, 16, 32);

    // C/D layout: vgpr r = row r (lanes 0-15) / row r+8 (lanes 16-31)
#pragma unroll
    for (int r = 0; r < 8; ++r) {
        float Zr  = __shfl(zfull, r + 8 * half, 32);
        float inv = 1.0f / Zr;
#pragma unroll
        for (int t = 0; t < 16; ++t) c[t][r] *= inv;
    }
#pragma unroll
    for (int t = 0; t < 16; ++t)
#pragma unroll
        for (int r = 0; r < 8; ++r)
            out[(size_t)(i0 + r + 8 * half) * OUT_DIM + t * 16 + m] = c[t][r];
}

// ---------------------------------------------------------------------------
extern "C" void kernel_launch(void* const* d_in, const int* in_sizes, int n_in,
                              void* d_out, int out_size, void* d_ws, size_t ws_size,
                              hipStream_t stream) {
    const float* x  = (const float*)d_in[0];
    // d_in[1] = adj  (accepted but unused by the reference module)
    const float* W  = (const float*)d_in[2];
    const float* a1 = (const float*)d_in[3];
    const float* a2 = (const float*)d_in[4];
    float* out = (float*)d_out;

    float* h   = (float*)d_ws;                     // 8192*256 f32 = 8 MB
    float* s1  = h + (size_t)N_ROWS * OUT_DIM;     // 8192 f32
    float* s2  = s1 + N_ROWS;                      // 8192 f32
    float* s1m = s2 + N_ROWS;                      // 1 f32

    gat_gemm_h<<<N_ROWS / 32, 256, 0, stream>>>(x, W, h);
    gat_s12<<<N_ROWS / 8, 256, 0, stream>>>(h, a1, a2, s1, s2);
    gat_s1max<<<1, 256, 0, stream>>>(s1, s1m);
    gat_attn<<<N_ROWS / 128, 256, 0, stream>>>(h, s1, s2, s1m, out);
}